// weighted_GCN_5927054868790
// MI455X (gfx1250) — compile-verified
//
#include <hip/hip_runtime.h>

// Problem constants (match reference)
#define N_NODES 10000
#define E_EDGES 320000
#define T_TYPES 8
#define F_FEAT  32
#define NTF     (N_NODES * T_TYPES * F_FEAT)   // 2,560,000 floats
#define MROWS   (N_NODES * T_TYPES)            // 80,000 rows for the GEMM
#define MTILES  (MROWS / 16)                   // 5000 16-row tiles
#define EPSV    1e-5f

typedef float v2f __attribute__((ext_vector_type(2)));
typedef float v8f __attribute__((ext_vector_type(8)));

// ---------------------------------------------------------------- zero
__global__ void zero_kernel(float* __restrict__ p, int n) {
  int i = blockIdx.x * blockDim.x + threadIdx.x;
  if (i < n) p[i] = 0.0f;
}

// ---------------------------------------------------------------- scatter
// One thread per (edge, f). Lanes of a wave span f for a fixed edge, so the
// 8 atomicAdds per t hit 32 consecutive floats (coalesced L2 atomics).
__global__ void scatter_kernel(const float* __restrict__ h,
                               const float* __restrict__ ew,   // (T, E)
                               const int*   __restrict__ src,
                               const int*   __restrict__ dst,
                               float*       __restrict__ agg,  // (N, T, F)
                               int firstLayer) {
  int tid = blockIdx.x * blockDim.x + threadIdx.x;
  int e = tid >> 5;
  int f = tid & 31;
  if (e >= E_EDGES) return;
  int s = src[e];
  int d = dst[e];
  float* ap = agg + (size_t)d * (T_TYPES * F_FEAT) + f;
  if (firstLayer) {
    // h is (N, F): broadcast one feature vector over T types
    float hv = h[(size_t)s * F_FEAT + f];
#pragma unroll
    for (int t = 0; t < T_TYPES; ++t)
      atomicAdd(ap + t * F_FEAT, ew[(size_t)t * E_EDGES + e] * hv);
  } else {
    // h is (N, T, F)
    const float* hp = h + (size_t)s * (T_TYPES * F_FEAT) + f;
#pragma unroll
    for (int t = 0; t < T_TYPES; ++t)
      atomicAdd(ap + t * F_FEAT, ew[(size_t)t * E_EDGES + e] * hp[t * F_FEAT]);
  }
}

// ---------------------------------------------------------------- GEMM + stats
// out[m][o] = sum_f A[m][f] * W[o][f] + b[o], computed with fp32 WMMA
// V_WMMA_F32_16X16X4_F32 (K=4), 8 k-steps x 2 column tiles per 16-row tile.
// Runs IN PLACE on AP (all A loads of a tile precede its stores).
// Also accumulates per-column sum / sum^2 for the layernorm.
__global__ void gemm_stats_kernel(const float* __restrict__ W,    // 32x32 [o][f]
                                  const float* __restrict__ bvec, // 32
                                  float* AP,                      // MROWS x 32 (in/out)
                                  float* __restrict__ colsum,     // 32
                                  float* __restrict__ colsq) {    // 32
  const int lane = threadIdx.x & 31;
  const int l16  = lane & 15;   // M (for A) / N (for B,D)
  const int half = lane >> 4;   // selects K pair (A,B) / M+8 (D)
  const int wavesPerBlk = blockDim.x >> 5;
  const int gwave  = blockIdx.x * wavesPerBlk + (threadIdx.x >> 5);
  const int nwaves = gridDim.x * wavesPerBlk;

  // B fragments: B[k][n] = W[n][k]. VGPR0 = B[4kk+2*half][n], VGPR1 = next k.
  v2f Bf[2][8];
#pragma unroll
  for (int nb = 0; nb < 2; ++nb)
#pragma unroll
    for (int kk = 0; kk < 8; ++kk)
      Bf[nb][kk] = *(const v2f*)(W + (nb * 16 + l16) * 32 + kk * 4 + half * 2);

  const float bias0 = bvec[l16];
  const float bias1 = bvec[16 + l16];

  float s0 = 0.f, q0 = 0.f, s1 = 0.f, q1 = 0.f;

  for (int tile = gwave; tile < MTILES; tile += nwaves) {
    const float* Ap = AP + (size_t)tile * 16 * 32;
    v2f Af[8];
#pragma unroll
    for (int kk = 0; kk < 8; ++kk)
      Af[kk] = *(const v2f*)(Ap + l16 * 32 + kk * 4 + half * 2);

    v8f acc0 = {0.f, 0.f, 0.f, 0.f, 0.f, 0.f, 0.f, 0.f};
    v8f acc1 = {0.f, 0.f, 0.f, 0.f, 0.f, 0.f, 0.f, 0.f};
#pragma unroll
    for (int kk = 0; kk < 8; ++kk) {
      acc0 = __builtin_amdgcn_wmma_f32_16x16x4_f32(
          false, Af[kk], false, Bf[0][kk], (short)0, acc0, false, false);
      acc1 = __builtin_amdgcn_wmma_f32_16x16x4_f32(
          false, Af[kk], false, Bf[1][kk], (short)0, acc1, false, false);
    }

    float* Pp = AP + (size_t)tile * 16 * 32;
#pragma unroll
    for (int r = 0; r < 8; ++r) {
      int row = r + half * 8;                 // D layout: M = r + 8*(lane/16)
      float v0 = acc0[r] + bias0;
      float v1 = acc1[r] + bias1;
      Pp[row * 32 + l16]      = v0;
      Pp[row * 32 + 16 + l16] = v1;
      s0 += v0; q0 += v0 * v0;
      s1 += v1; q1 += v1 * v1;
    }
  }

  // lane L and L+16 hold the same output column -> fold halves (wave32)
  s0 += __shfl_xor(s0, 16, 32);
  q0 += __shfl_xor(q0, 16, 32);
  s1 += __shfl_xor(s1, 16, 32);
  q1 += __shfl_xor(q1, 16, 32);
  if (half == 0) {
    atomicAdd(&colsum[l16],      s0);
    atomicAdd(&colsq[l16],       q0);
    atomicAdd(&colsum[16 + l16], s1);
    atomicAdd(&colsq[16 + l16],  q1);
  }
}

// ---------------------------------------------------------------- finalize stats
__global__ void finalize_kernel(const float* __restrict__ colsum,
                                const float* __restrict__ colsq,
                                const float* __restrict__ gamma,
                                const float* __restrict__ beta,
                                float* __restrict__ scale,
                                float* __restrict__ shift) {
  int c = threadIdx.x;
  if (c < 32) {
    const float inv  = 1.0f / (float)MROWS;
    float mean = colsum[c] * inv;
    float var  = colsq[c] * inv - mean * mean;
    float sc   = gamma[c] * rsqrtf(var + EPSV);
    scale[c] = sc;
    shift[c] = beta[c] - mean * sc;
  }
}

// ---------------------------------------------------------------- normalize + relu
__global__ void norm_relu_kernel(const float* __restrict__ pre,
                                 const float* __restrict__ scale,
                                 const float* __restrict__ shift,
                                 float* __restrict__ outp) {
  int i = blockIdx.x * blockDim.x + threadIdx.x;
  if (i < NTF) {
    int c = i & 31;
    float v = pre[i] * scale[c] + shift[c];
    outp[i] = v > 0.f ? v : 0.f;
  }
}

// ---------------------------------------------------------------- launcher
extern "C" void kernel_launch(void* const* d_in, const int* in_sizes, int n_in,
                              void* d_out, int out_size, void* d_ws, size_t ws_size,
                              hipStream_t stream) {
  const float* nodef  = (const float*)d_in[0];  // (N, F)
  const float* ew     = (const float*)d_in[1];  // (T, E)
  const int*   src    = (const int*)  d_in[2];  // (E)
  const int*   dst    = (const int*)  d_in[3];  // (E)
  const float* Ws     = (const float*)d_in[4];  // (L, F, F)
  const float* bs     = (const float*)d_in[5];  // (L, F)
  const float* gammas = (const float*)d_in[6];  // (L, F)
  const float* betas  = (const float*)d_in[7];  // (L, F)
  float* out = (float*)d_out;                   // (N, T, F)

  float* ws    = (float*)d_ws;
  float* agg   = ws;              // NTF floats: agg, then in-place pre-norm out
  float* hbuf  = ws + NTF;        // NTF floats: h between layers
  float* stats = ws + 2 * (size_t)NTF; // sum[32] | sumsq[32] | scale[32] | shift[32]
  float* colsum = stats;
  float* colsq  = stats + 32;
  float* scale  = stats + 64;
  float* shift  = stats + 96;

  const int ZB = (NTF + 255) / 256;            // 10000 blocks
  const int SB = (E_EDGES * 32 + 255) / 256;   // 40000 blocks

  const float* hcur = nodef;
  for (int l = 0; l < 3; ++l) {
    zero_kernel<<<ZB, 256, 0, stream>>>(agg, NTF);
    zero_kernel<<<1, 64, 0, stream>>>(stats, 64);

    scatter_kernel<<<SB, 256, 0, stream>>>(hcur, ew, src, dst, agg, l == 0 ? 1 : 0);

    gemm_stats_kernel<<<128, 256, 0, stream>>>(Ws + (size_t)l * F_FEAT * F_FEAT,
                                               bs + l * F_FEAT,
                                               agg, colsum, colsq);

    finalize_kernel<<<1, 32, 0, stream>>>(colsum, colsq,
                                          gammas + l * F_FEAT, betas + l * F_FEAT,
                                          scale, shift);

    float* hnext = (l == 2) ? out : hbuf;
    norm_relu_kernel<<<ZB, 256, 0, stream>>>(agg, scale, shift, hnext);
    hcur = hnext;
  }
}